// RNN_basic_73993696575507
// MI455X (gfx1250) — compile-verified
//
#include <hip/hip_runtime.h>
#include <math.h>

typedef __attribute__((ext_vector_type(2))) float v2f;
typedef __attribute__((ext_vector_type(8))) float v8f;
typedef __attribute__((ext_vector_type(4))) unsigned int u32x4;
typedef __attribute__((ext_vector_type(8))) int i32x8;
typedef __attribute__((ext_vector_type(4))) int i32x4;

#define TSTEPS 500
#define NB     16        // batch
#define HID    512       // H
#define GG     1024      // 2H
#define MROWS  (TSTEPS * 32)   // 16000

#if __has_builtin(__builtin_amdgcn_tensor_load_to_lds)
#define HAVE_TDM 1
#else
#define HAVE_TDM 0
#endif

// ---------------------------------------------------------------------------
// Utility: zero a float region
// ---------------------------------------------------------------------------
__global__ void zero_f32(float* __restrict__ p, int n) {
    int i = blockIdx.x * blockDim.x + threadIdx.x;
    if (i < n) p[i] = 0.0f;
}

// ---------------------------------------------------------------------------
// Pre-GEMM: w[m, g] = sum_f x2[m, f] * Ww[g, f]
//   m = t*32 + r ; r<16 -> forward row, r>=16 -> time-reversed row.
//   layer 0: X is (B=16, T, F) ; layer 1: X is (T, B=16, 2H)
//   Each wave computes one 16x16 tile using V_WMMA_F32_16X16X4_F32.
// ---------------------------------------------------------------------------
__global__ void pregemm_wmma(const float* __restrict__ X, const float* __restrict__ Ww,
                             float* __restrict__ wbuf, int K, int layer) {
    const int lane = threadIdx.x & 31;
    const int wave = threadIdx.x >> 5;
    const int tile = blockIdx.x * 4 + wave;   // 0 .. 63999
    const int nt = tile & 63;                 // 64 N-tiles of 16
    const int mt = tile >> 6;                 // 1000 M-tiles of 16
    const int m_base = mt * 16;
    const int t = m_base >> 5;                // time index (tile fits in one t)
    const int r_base = m_base & 31;           // 0 or 16
    const int r = r_base + (lane & 15);       // row within the 32 "batch" rows

    const float* aptr;
    if (layer == 0) {
        // x (16, T, F): forward b=r at time t; backward b=r-16 at time T-1-t
        aptr = (r < NB) ? X + ((size_t)r * TSTEPS + t) * (size_t)K
                        : X + ((size_t)(r - NB) * TSTEPS + (TSTEPS - 1 - t)) * (size_t)K;
    } else {
        // x1 (T, 16, 2H)
        aptr = (r < NB) ? X + ((size_t)t * NB + r) * (size_t)K
                        : X + ((size_t)(TSTEPS - 1 - t) * NB + (r - NB)) * (size_t)K;
    }
    const int koff = (lane >> 4) * 2;         // K sub-offset per lane half
    aptr += koff;
    const int g = nt * 16 + (lane & 15);      // output feature column
    const float* bptr = Ww + (size_t)g * (size_t)K + koff;

    v8f c = {};
    for (int kb = 0; kb < K; kb += 4) {
        v2f af = *(const v2f*)(aptr + kb);
        v2f bf = *(const v2f*)(bptr + kb);
        c = __builtin_amdgcn_wmma_f32_16x16x4_f32(false, af, false, bf,
                                                  (short)0, c, false, false);
    }
    const int rowadd = (lane >= 16) ? 8 : 0;
#pragma unroll
    for (int i = 0; i < 8; ++i) {
        int m = m_base + i + rowadd;
        wbuf[(size_t)m * GG + g] = c[i];
    }
}

// ---------------------------------------------------------------------------
// LayerNorm stats: per-feature sum / sumsq over all 16000 rows.
// ---------------------------------------------------------------------------
__global__ void ln_stats(const float* __restrict__ w, float* __restrict__ sum,
                         float* __restrict__ sumsq) {
    const int f0 = threadIdx.x * 4;
    const size_t rowbase = (size_t)blockIdx.x * 64;
    float s[4] = {0, 0, 0, 0}, s2[4] = {0, 0, 0, 0};
    for (int r = 0; r < 64; ++r) {
        const float* row = w + (rowbase + r) * GG + f0;
#pragma unroll
        for (int i = 0; i < 4; ++i) { float v = row[i]; s[i] += v; s2[i] += v * v; }
    }
#pragma unroll
    for (int i = 0; i < 4; ++i) {
        atomicAdd(&sum[f0 + i], s[i]);
        atomicAdd(&sumsq[f0 + i], s2[i]);
    }
}

// scale[g] = gamma * rsqrt(var + eps) ; shift[g] = beta - mean*scale
__global__ void ln_final(const float* __restrict__ sum, const float* __restrict__ sumsq,
                         const float* __restrict__ gamma, const float* __restrict__ beta,
                         float* __restrict__ scale, float* __restrict__ shift) {
    int g = blockIdx.x * blockDim.x + threadIdx.x;
    if (g < GG) {
        const float inv = 1.0f / (float)MROWS;
        float m = sum[g] * inv;
        float v = sumsq[g] * inv - m * m;
        float sc = gamma[g] * rsqrtf(v + 1e-5f);
        scale[g] = sc;
        shift[g] = beta[g] - m * sc;
    }
}

// ---------------------------------------------------------------------------
// TDM: load a (nrows x rowlen) f32 row-block Wu[g0..g0+nrows) into LDS.
// D# built per CDNA5 ISA (2D, data_size=4B).
// 6-arg builtin variant (clang-23 / therock-10.0 toolchain).
// ---------------------------------------------------------------------------
#if HAVE_TDM
__device__ __forceinline__ void tdm_load_rows_f32(const float* gptr, unsigned lds_off,
                                                  unsigned nrows, unsigned rowlen) {
    unsigned long long ga = (unsigned long long)(size_t)gptr;
    u32x4 g0;
    g0[0] = 1u;                                          // count=1, user mode
    g0[1] = lds_off;                                     // lds_addr (bytes)
    g0[2] = (unsigned)(ga & 0xFFFFFFFFu);                // global_addr[31:0]
    g0[3] = (unsigned)((ga >> 32) & 0x1FFFFFFu) | (2u << 30); // global_addr[56:32], type=2
    // group1 as 4 x 64-bit quadwords (bits 255:0)
    unsigned long long q0 = ((unsigned long long)2u << 16)                    // data_size=4B
                          | ((unsigned long long)(rowlen & 0xFFFFu) << 48);   // tensor_dim0[15:0]
    unsigned long long q1 = (unsigned long long)((rowlen >> 16) & 0xFFFFu)    // tensor_dim0[31:16]
                          | ((unsigned long long)nrows << 16)                 // tensor_dim1
                          | ((unsigned long long)(rowlen & 0xFFFFu) << 48);   // tile_dim0
    unsigned long long q2 = (unsigned long long)(nrows & 0xFFFFu)             // tile_dim1
                          | ((unsigned long long)rowlen << 32);               // tensor_dim0_stride[31:0]
    unsigned long long q3 = 0ull;                                             // stride[47:32]=0, dim1_stride=0
    i32x8 g1;
    g1[0] = (int)(unsigned)(q0 & 0xFFFFFFFFu); g1[1] = (int)(unsigned)(q0 >> 32);
    g1[2] = (int)(unsigned)(q1 & 0xFFFFFFFFu); g1[3] = (int)(unsigned)(q1 >> 32);
    g1[4] = (int)(unsigned)(q2 & 0xFFFFFFFFu); g1[5] = (int)(unsigned)(q2 >> 32);
    g1[6] = (int)(unsigned)(q3 & 0xFFFFFFFFu); g1[7] = (int)(unsigned)(q3 >> 32);
    i32x4 gz4 = {0, 0, 0, 0};
    i32x8 gz8 = {0, 0, 0, 0, 0, 0, 0, 0};
    __builtin_amdgcn_tensor_load_to_lds(g0, g1, gz4, gz4, gz8, 0);
}
#endif

// ---------------------------------------------------------------------------
// Persistent Li-GRU recurrence. 16 WGs x 128 threads (4 waves).
// Dynamic LDS: wu_s[64*512] (128K) | h_s[32*512] (64K) | a_s[1K] | z_s[1K]
// Wu slice is TDM-loaded to LDS once; h_prev staged to LDS each step.
// ---------------------------------------------------------------------------
__global__ void ligru_recurrence(const float* __restrict__ wbuf,
                                 const float* __restrict__ scale,
                                 const float* __restrict__ shift,
                                 const float* __restrict__ Wu,
                                 float* __restrict__ hping, float* __restrict__ hpong,
                                 float* __restrict__ hall, int* __restrict__ ctr) {
    extern __shared__ float smem[];
    float* wu_s = smem;                  // 64 rows x 512
    float* h_s  = wu_s + 64 * HID;       // 32 rows x 512
    float* a_s  = h_s + 32 * HID;        // 32 x 32
    float* z_s  = a_s + 1024;            // 32 x 32

    const int wg   = blockIdx.x;         // 0..15
    const int j0   = wg * 32;            // H-column base
    const int wave = threadIdx.x >> 5;   // 0..3
    const int lane = threadIdx.x & 31;
    const int is_z = wave >> 1;          // waves 2,3 compute the z-gate slice
    const int nh   = wave & 1;           // which 16-col half
    const int g    = is_z * HID + j0 + nh * 16 + (lane & 15);  // gate column
    const int koff = (lane >> 4) * 2;
    const int col  = nh * 16 + (lane & 15);
    const float sc = scale[g];
    const float sh = shift[g];
    const int rowadd = (lane >= 16) ? 8 : 0;

    // ---- one-time Wu slice -> LDS (rows [j0,j0+32) then [512+j0, 512+j0+32)) ----
#if HAVE_TDM
    if (wave == 0) {
        unsigned wu_off = (unsigned)(size_t)(void*)wu_s;   // flat addr low 32b == LDS offset
        tdm_load_rows_f32(Wu + (size_t)j0 * HID,         wu_off,                  32u, (unsigned)HID);
        tdm_load_rows_f32(Wu + (size_t)(HID + j0) * HID, wu_off + 32u * HID * 4u, 32u, (unsigned)HID);
        __builtin_amdgcn_s_wait_tensorcnt(0);
    }
#else
    for (int e = threadIdx.x; e < 64 * HID / 4; e += 128) {
        int row = (e * 4) / HID;                 // 0..63 local row
        int k   = (e * 4) % HID;
        int grow = (row < 32) ? (j0 + row) : (HID + j0 + (row - 32));
        *(float4*)(wu_s + row * HID + k) = *(const float4*)(Wu + (size_t)grow * HID + k);
    }
#endif
    __syncthreads();

    // LDS row of Wu this wave multiplies (B-matrix fragment source)
    const float* bls = wu_s + (size_t)(is_z * 32 + nh * 16 + (lane & 15)) * HID + koff;
    const float* a0l = h_s + (size_t)(lane & 15) * HID + koff;
    const float* a1l = h_s + (size_t)(16 + (lane & 15)) * HID + koff;

    for (int t = 0; t < TSTEPS; ++t) {
        const float* hprev = (t & 1) ? hpong : hping;
        float*       hnext = (t & 1) ? hping : hpong;

        // stage h_prev (32x512) into LDS, shared by all 4 waves
        for (int e = threadIdx.x; e < 32 * HID / 4; e += 128) {
            *(float4*)(h_s + e * 4) = *(const float4*)(hprev + e * 4);
        }
        __syncthreads();

        // gates tile = h_prev(32x512) @ Wu^T slice, two 16-row halves (f32 WMMA)
        v8f c0 = {}, c1 = {};
        for (int kb = 0; kb < HID; kb += 4) {
            v2f bf  = *(const v2f*)(bls + kb);
            v2f af0 = *(const v2f*)(a0l + kb);
            v2f af1 = *(const v2f*)(a1l + kb);
            c0 = __builtin_amdgcn_wmma_f32_16x16x4_f32(false, af0, false, bf,
                                                       (short)0, c0, false, false);
            c1 = __builtin_amdgcn_wmma_f32_16x16x4_f32(false, af1, false, bf,
                                                       (short)0, c1, false, false);
        }

        // add layer-normed w[t] and stage gates in LDS
        float* sbuf = is_z ? z_s : a_s;
        const float* wrow = wbuf + (size_t)t * 32 * GG;
#pragma unroll
        for (int i = 0; i < 8; ++i) {
            int r0 = i + rowadd;
            sbuf[r0 * 32 + col] = c0[i] + wrow[(size_t)r0 * GG + g] * sc + sh;
            int r1 = 16 + r0;
            sbuf[r1 * 32 + col] = c1[i] + wrow[(size_t)r1 * GG + g] * sc + sh;
        }
        __syncthreads();

        // gate update: 32 rows x 32 cols over 128 threads
        for (int e = threadIdx.x; e < 1024; e += 128) {
            int r  = e >> 5;
            int cc = e & 31;
            float a  = a_s[r * 32 + cc];
            float z  = z_s[r * 32 + cc];
            float zs = 1.0f / (1.0f + __expf(-z));
            float hc = a > 0.0f ? a : 0.0f;
            float hp = h_s[(size_t)r * HID + j0 + cc];
            float hn = zs * hp + (1.0f - zs) * hc;
            hnext[(size_t)r * HID + j0 + cc] = hn;
            hall[((size_t)t * 32 + r) * HID + j0 + cc] = hn;
        }
        __syncthreads();

        // device-wide barrier (monotonic counter, sleep-spin)
        if (threadIdx.x == 0) {
            __threadfence();
            atomicAdd(ctr, 1);
            const int target = 16 * (t + 1);
            while (atomicAdd(ctr, 0) < target) { __builtin_amdgcn_s_sleep(1); }
            __threadfence();
        }
        __syncthreads();
    }
}

// ---------------------------------------------------------------------------
// x1[t,b,j] = j<H ? h0[t, b, j] : h0[T-1-t, 16+b, j-H]     (T,16,1024)
// ---------------------------------------------------------------------------
__global__ void make_x1(const float* __restrict__ hall, float* __restrict__ x1) {
    size_t idx = (size_t)blockIdx.x * blockDim.x + threadIdx.x;
    if (idx >= (size_t)TSTEPS * NB * GG) return;
    int j = (int)(idx & (GG - 1));
    size_t rem = idx >> 10;
    int b = (int)(rem & (NB - 1));
    int t = (int)(rem >> 4);
    float v;
    if (j < HID) v = hall[((size_t)t * 32 + b) * HID + j];
    else         v = hall[((size_t)(TSTEPS - 1 - t) * 32 + NB + b) * HID + (j - HID)];
    x1[idx] = v;
}

// out (B,T,2H): out[b,t,j] = j<H ? h1[t, b, j] : h1[T-1-t, 16+b, j-H]
__global__ void final_out(const float* __restrict__ hall, float* __restrict__ out) {
    size_t idx = (size_t)blockIdx.x * blockDim.x + threadIdx.x;
    if (idx >= (size_t)NB * TSTEPS * GG) return;
    int j = (int)(idx & (GG - 1));
    size_t rem = idx >> 10;
    int t = (int)(rem % TSTEPS);
    int b = (int)(rem / TSTEPS);
    float v;
    if (j < HID) v = hall[((size_t)t * 32 + b) * HID + j];
    else         v = hall[((size_t)(TSTEPS - 1 - t) * 32 + NB + b) * HID + (j - HID)];
    out[idx] = v;
}

// ---------------------------------------------------------------------------
extern "C" void kernel_launch(void* const* d_in, const int* in_sizes, int n_in,
                              void* d_out, int out_size, void* d_ws, size_t ws_size,
                              hipStream_t stream) {
    (void)in_sizes; (void)n_in; (void)out_size; (void)ws_size;
    const float* x   = (const float*)d_in[0];
    const float* Ww0 = (const float*)d_in[1];
    const float* Wu0 = (const float*)d_in[2];
    const float* g0  = (const float*)d_in[3];
    const float* b0  = (const float*)d_in[4];
    const float* Ww1 = (const float*)d_in[5];
    const float* Wu1 = (const float*)d_in[6];
    const float* g1  = (const float*)d_in[7];
    const float* b1  = (const float*)d_in[8];
    float* out = (float*)d_out;

    // workspace layout (floats): ~131 MB total
    float* ws    = (float*)d_ws;
    float* wbuf  = ws;                               // 16,384,000  gates (reused per layer)
    float* hall  = wbuf + (size_t)16384000;          //  8,192,000  h_all (reused per layer)
    float* x1    = hall + (size_t)8192000;           //  8,192,000  layer-1 input
    float* sum   = x1   + (size_t)8192000;           //  1024
    float* sumsq = sum + 1024;                       //  1024
    float* scale = sumsq + 1024;                     //  1024
    float* shift = scale + 1024;                     //  1024
    float* hping = shift + 1024;                     //  16384
    float* hpong = hping + 16384;                    //  16384
    int*   ctr   = (int*)(hpong + 16384);            //  1 (zeroed as float bits)

    // dynamic LDS for recurrence: Wu slice 128K + h 64K + gates 8K = 200K (<320K/WG)
    const size_t rec_lds = (size_t)(64 * HID + 32 * HID + 2048) * sizeof(float);

    auto run_layer = [&](const float* Xin, const float* Ww, const float* Wu,
                         const float* gamma, const float* beta, int K, int layer) {
        zero_f32<<<(2048 + 255) / 256, 256, 0, stream>>>(sum, 2048);       // sum+sumsq
        zero_f32<<<(32769 + 255) / 256, 256, 0, stream>>>(hping, 32769);   // h0=0, ctr=0
        pregemm_wmma<<<16000, 128, 0, stream>>>(Xin, Ww, wbuf, K, layer);
        ln_stats<<<250, 256, 0, stream>>>(wbuf, sum, sumsq);
        ln_final<<<4, 256, 0, stream>>>(sum, sumsq, gamma, beta, scale, shift);
        ligru_recurrence<<<16, 128, rec_lds, stream>>>(wbuf, scale, shift, Wu,
                                                       hping, hpong, hall, ctr);
    };

    run_layer(x, Ww0, Wu0, g0, b0, 240, 0);
    make_x1<<<32000, 256, 0, stream>>>(hall, x1);
    run_layer(x1, Ww1, Wu1, g1, b1, GG, 1);
    final_out<<<32000, 256, 0, stream>>>(hall, out);
}